// GCN_54331336294861
// MI455X (gfx1250) — compile-verified
//
#include <hip/hip_runtime.h>

#define DIM   256
#define LN_EPS 1e-5f

typedef __attribute__((ext_vector_type(2))) float v2f;
typedef __attribute__((ext_vector_type(8))) float v8f;

// ---------------------------------------------------------------------------
// Utility: fill a float buffer with a constant (used to init deg=1, zero sums)
// ---------------------------------------------------------------------------
__global__ __launch_bounds__(256) void k_fill(float* __restrict__ p, float v, int n) {
  int i = blockIdx.x * 256 + threadIdx.x;
  if (i < n) p[i] = v;
}

// ---------------------------------------------------------------------------
// deg[dst] += 1 per edge (deg pre-initialized to 1.0 for the self-loop)
// ---------------------------------------------------------------------------
__global__ __launch_bounds__(256) void k_degree(const int* __restrict__ dst,
                                                float* __restrict__ deg, int e) {
  int i = blockIdx.x * 256 + threadIdx.x;
  if (i < e) atomicAdd(&deg[dst[i]], 1.0f);
}

__global__ __launch_bounds__(256) void k_dinv(const float* __restrict__ deg,
                                              float* __restrict__ dinv, int n) {
  int i = blockIdx.x * 256 + threadIdx.x;
  if (i < n) dinv[i] = rsqrtf(deg[i]);
}

// ---------------------------------------------------------------------------
// C[M,256] = A[M,256] @ W[256,256] using V_WMMA_F32_16X16X4_F32.
// One block per 16-row stripe of A; 8 waves each compute two 16x16 tiles of C
// (reusing the A fragment for both, halving A traffic).
// ---------------------------------------------------------------------------
__global__ __launch_bounds__(256) void gemm_xw(const float* __restrict__ A,
                                               const float* __restrict__ W,
                                               float* __restrict__ C, int M) {
  const int lane  = threadIdx.x & 31;
  const int wave  = threadIdx.x >> 5;
  const int mBase = blockIdx.x * 16;
  if (mBase >= M) return;
  const int n0   = wave * 32;                 // this wave's two N-tiles: n0, n0+16
  const int kOff = (lane >> 4) * 2;           // ISA A/B layout: upper half-wave holds K+2..K+3
  const int nCol = lane & 15;
  int mRow = mBase + (lane & 15);
  if (mRow >= M) mRow = M - 1;                // safe clamp (N % 16 == 0 in practice)

  v8f c0 = {}; v8f c1 = {};
  for (int k = 0; k < DIM; k += 4) {
    v2f a, b0, b1;
    a.x  = A[(size_t)mRow * DIM + k + kOff];
    a.y  = A[(size_t)mRow * DIM + k + kOff + 1];
    b0.x = W[(size_t)(k + kOff) * DIM + n0 + nCol];
    b0.y = W[(size_t)(k + kOff + 1) * DIM + n0 + nCol];
    b1.x = W[(size_t)(k + kOff) * DIM + n0 + 16 + nCol];
    b1.y = W[(size_t)(k + kOff + 1) * DIM + n0 + 16 + nCol];
    c0 = __builtin_amdgcn_wmma_f32_16x16x4_f32(false, a, false, b0, (short)0, c0, false, false);
    c1 = __builtin_amdgcn_wmma_f32_16x16x4_f32(false, a, false, b1, (short)0, c1, false, false);
  }
#pragma unroll
  for (int i = 0; i < 8; ++i) {
    int row = mBase + i + (lane >> 4) * 8;    // ISA C/D layout
    if (row < M) {
      C[(size_t)row * DIM + n0 + nCol]      = c0[i];
      C[(size_t)row * DIM + n0 + 16 + nCol] = c1[i];
    }
  }
}

// ---------------------------------------------------------------------------
// agg[i,:] = hw[i,:] * dinv[i]^2 + bias[:]   (self-loop term + conv bias;
// also serves as the zero-init of agg before the edge scatter)
// One wave per row; fully coalesced float4 path.
// ---------------------------------------------------------------------------
__global__ __launch_bounds__(256) void k_selfloop(const float* __restrict__ hw,
                                                  const float* __restrict__ dinv,
                                                  const float* __restrict__ bias,
                                                  float* __restrict__ agg, int n) {
  int row  = blockIdx.x * 8 + (threadIdx.x >> 5);
  int lane = threadIdx.x & 31;
  if (row >= n) return;
  float inv = dinv[row]; inv *= inv;          // 1/deg
  const float4* src = (const float4*)(hw  + (size_t)row * DIM);
  float4*       dst = (float4*)(agg + (size_t)row * DIM);
  const float4* b4  = (const float4*)bias;
  float4 v0 = src[lane], v1 = src[lane + 32];
  float4 bb0 = b4[lane], bb1 = b4[lane + 32];
  dst[lane]      = make_float4(v0.x * inv + bb0.x, v0.y * inv + bb0.y,
                               v0.z * inv + bb0.z, v0.w * inv + bb0.w);
  dst[lane + 32] = make_float4(v1.x * inv + bb1.x, v1.y * inv + bb1.y,
                               v1.z * inv + bb1.z, v1.w * inv + bb1.w);
}

// ---------------------------------------------------------------------------
// Edge scatter: agg[dst,:] += hw[src,:] * (dinv[src]*dinv[dst]).
// One wave per edge; b128 gathers, non-returning f32 atomics (STOREcnt path,
// resolve at L2 since agg (51 MB) is L2-resident). Prefetch next edge's row.
// ---------------------------------------------------------------------------
__global__ __launch_bounds__(256) void k_scatter(const int* __restrict__ src,
                                                 const int* __restrict__ dst,
                                                 const float* __restrict__ dinv,
                                                 const float* __restrict__ hw,
                                                 float* __restrict__ agg, int e) {
  int edge = blockIdx.x * 8 + (threadIdx.x >> 5);
  int lane = threadIdx.x & 31;
  if (edge >= e) return;
  int s = src[edge], d = dst[edge];
  float w = dinv[s] * dinv[d];

  if (edge + 8 < e) {                         // prefetch next edge's source row
    int s2 = src[edge + 8];
    __builtin_prefetch(hw + (size_t)s2 * DIM + lane * 8, 0, 0);
  }

  const float4* hs = (const float4*)(hw + (size_t)s * DIM);
  float*        ad = agg + (size_t)d * DIM;
  float4 v0 = hs[lane];                       // cols lane*4 .. lane*4+3
  float4 v1 = hs[lane + 32];                  // cols 128+lane*4 ..
  int c0 = lane * 4, c1 = 128 + lane * 4;
  atomicAdd(&ad[c0 + 0], v0.x * w);
  atomicAdd(&ad[c0 + 1], v0.y * w);
  atomicAdd(&ad[c0 + 2], v0.z * w);
  atomicAdd(&ad[c0 + 3], v0.w * w);
  atomicAdd(&ad[c1 + 0], v1.x * w);
  atomicAdd(&ad[c1 + 1], v1.y * w);
  atomicAdd(&ad[c1 + 2], v1.z * w);
  atomicAdd(&ad[c1 + 3], v1.w * w);
}

// ---------------------------------------------------------------------------
// Fused LayerNorm + ReLU, in place. One wave32 per 256-wide row; shfl_xor
// tree reductions (width 32 — gfx1250 is wave32).
// ---------------------------------------------------------------------------
__global__ __launch_bounds__(256) void k_ln_relu(float* __restrict__ h,
                                                 const float* __restrict__ g,
                                                 const float* __restrict__ be, int n) {
  int row  = blockIdx.x * 8 + (threadIdx.x >> 5);
  int lane = threadIdx.x & 31;
  if (row >= n) return;
  float4* p = (float4*)(h + (size_t)row * DIM);
  float4 v0 = p[lane], v1 = p[lane + 32];

  float s = v0.x + v0.y + v0.z + v0.w + v1.x + v1.y + v1.z + v1.w;
#pragma unroll
  for (int off = 16; off >= 1; off >>= 1) s += __shfl_xor(s, off, 32);
  float mu = s * (1.0f / DIM);

  float dx, var = 0.0f;
  dx = v0.x - mu; var += dx * dx;  dx = v0.y - mu; var += dx * dx;
  dx = v0.z - mu; var += dx * dx;  dx = v0.w - mu; var += dx * dx;
  dx = v1.x - mu; var += dx * dx;  dx = v1.y - mu; var += dx * dx;
  dx = v1.z - mu; var += dx * dx;  dx = v1.w - mu; var += dx * dx;
#pragma unroll
  for (int off = 16; off >= 1; off >>= 1) var += __shfl_xor(var, off, 32);
  float inv = rsqrtf(var * (1.0f / DIM) + LN_EPS);

  const float4* g4  = (const float4*)g;
  const float4* be4 = (const float4*)be;
  float4 gg0 = g4[lane], gg1 = g4[lane + 32];
  float4 bb0 = be4[lane], bb1 = be4[lane + 32];
  float4 o0, o1;
  o0.x = fmaxf((v0.x - mu) * inv * gg0.x + bb0.x, 0.0f);
  o0.y = fmaxf((v0.y - mu) * inv * gg0.y + bb0.y, 0.0f);
  o0.z = fmaxf((v0.z - mu) * inv * gg0.z + bb0.z, 0.0f);
  o0.w = fmaxf((v0.w - mu) * inv * gg0.w + bb0.w, 0.0f);
  o1.x = fmaxf((v1.x - mu) * inv * gg1.x + bb1.x, 0.0f);
  o1.y = fmaxf((v1.y - mu) * inv * gg1.y + bb1.y, 0.0f);
  o1.z = fmaxf((v1.z - mu) * inv * gg1.z + bb1.z, 0.0f);
  o1.w = fmaxf((v1.w - mu) * inv * gg1.w + bb1.w, 0.0f);
  p[lane] = o0; p[lane + 32] = o1;
}

// ---------------------------------------------------------------------------
// Global mean pool (accumulate): sums[batch[i],:] += h[i,:]; cnt[batch[i]]++
// ---------------------------------------------------------------------------
__global__ __launch_bounds__(256) void k_pool(const float* __restrict__ h,
                                              const int* __restrict__ batch,
                                              float* __restrict__ sums,
                                              float* __restrict__ cnt, int n) {
  int row  = blockIdx.x * 8 + (threadIdx.x >> 5);
  int lane = threadIdx.x & 31;
  if (row >= n) return;
  int g = batch[row];
  const float4* p = (const float4*)(h + (size_t)row * DIM);
  float* sg = sums + (size_t)g * DIM;
  float4 v0 = p[lane], v1 = p[lane + 32];
  int c0 = lane * 4, c1 = 128 + lane * 4;
  atomicAdd(&sg[c0 + 0], v0.x); atomicAdd(&sg[c0 + 1], v0.y);
  atomicAdd(&sg[c0 + 2], v0.z); atomicAdd(&sg[c0 + 3], v0.w);
  atomicAdd(&sg[c1 + 0], v1.x); atomicAdd(&sg[c1 + 1], v1.y);
  atomicAdd(&sg[c1 + 2], v1.z); atomicAdd(&sg[c1 + 3], v1.w);
  if (lane == 0) atomicAdd(&cnt[g], 1.0f);
}

__global__ __launch_bounds__(256) void k_pooled(const float* __restrict__ sums,
                                                const float* __restrict__ cnt,
                                                float* __restrict__ pooled, int total) {
  int i = blockIdx.x * 256 + threadIdx.x;
  if (i < total) {
    int g = i >> 8;                            // i / DIM
    pooled[i] = sums[i] / fmaxf(cnt[g], 1.0f);
  }
}

// ---------------------------------------------------------------------------
// Out[M,256] = relu(P[M,256] @ fc_w^T + fc_b) — WMMA with transposed B access.
// ---------------------------------------------------------------------------
__global__ __launch_bounds__(256) void gemm_fc(const float* __restrict__ P,
                                               const float* __restrict__ Wt,
                                               const float* __restrict__ bias,
                                               float* __restrict__ Out, int M) {
  const int lane  = threadIdx.x & 31;
  const int wave  = threadIdx.x >> 5;
  const int mBase = blockIdx.x * 16;
  if (mBase >= M) return;
  const int n0   = wave * 32;
  const int kOff = (lane >> 4) * 2;
  const int nCol = lane & 15;
  int mRow = mBase + (lane & 15);
  if (mRow >= M) mRow = M - 1;

  v8f c0 = {}; v8f c1 = {};
  for (int k = 0; k < DIM; k += 4) {
    v2f a, b0, b1;
    a.x  = P[(size_t)mRow * DIM + k + kOff];
    a.y  = P[(size_t)mRow * DIM + k + kOff + 1];
    // B[k][n] = fc_w[n][k]  (transposed read)
    b0.x = Wt[(size_t)(n0 + nCol) * DIM + k + kOff];
    b0.y = Wt[(size_t)(n0 + nCol) * DIM + k + kOff + 1];
    b1.x = Wt[(size_t)(n0 + 16 + nCol) * DIM + k + kOff];
    b1.y = Wt[(size_t)(n0 + 16 + nCol) * DIM + k + kOff + 1];
    c0 = __builtin_amdgcn_wmma_f32_16x16x4_f32(false, a, false, b0, (short)0, c0, false, false);
    c1 = __builtin_amdgcn_wmma_f32_16x16x4_f32(false, a, false, b1, (short)0, c1, false, false);
  }
  float bv0 = bias[n0 + nCol], bv1 = bias[n0 + 16 + nCol];
#pragma unroll
  for (int i = 0; i < 8; ++i) {
    int row = mBase + i + (lane >> 4) * 8;
    if (row < M) {
      Out[(size_t)row * DIM + n0 + nCol]      = fmaxf(c0[i] + bv0, 0.0f);
      Out[(size_t)row * DIM + n0 + 16 + nCol] = fmaxf(c1[i] + bv1, 0.0f);
    }
  }
}

// ---------------------------------------------------------------------------
extern "C" void kernel_launch(void* const* d_in, const int* in_sizes, int n_in,
                              void* d_out, int out_size, void* d_ws, size_t ws_size,
                              hipStream_t stream) {
  const float* x     = (const float*)d_in[0];
  const int*   ei    = (const int*)d_in[1];
  const int*   batch = (const int*)d_in[2];
  const float* W[3]  = { (const float*)d_in[3],  (const float*)d_in[7],  (const float*)d_in[11] };
  const float* b[3]  = { (const float*)d_in[4],  (const float*)d_in[8],  (const float*)d_in[12] };
  const float* g[3]  = { (const float*)d_in[5],  (const float*)d_in[9],  (const float*)d_in[13] };
  const float* be[3] = { (const float*)d_in[6],  (const float*)d_in[10], (const float*)d_in[14] };
  const float* fc_w  = (const float*)d_in[15];
  const float* fc_b  = (const float*)d_in[16];
  float* out = (float*)d_out;

  const int N = in_sizes[0] / DIM;
  const int E = in_sizes[1] / 2;
  const int G = out_size / DIM;
  const int* src = ei;
  const int* dst = ei + E;

  // Workspace carve-up
  char* ws = (char*)d_ws;
  auto carve = [&](size_t bytes) { char* p = ws; ws += (bytes + 255) & ~(size_t)255; return p; };
  float* bufA   = (float*)carve((size_t)N * DIM * 4);  // hw = h @ W
  float* bufB   = (float*)carve((size_t)N * DIM * 4);  // agg / layer output
  float* deg    = (float*)carve((size_t)N * 4);
  float* dinv   = (float*)carve((size_t)N * 4);
  float* sums   = (float*)carve((size_t)G * DIM * 4);
  float* cnt    = (float*)carve((size_t)G * 4);
  float* pooled = (float*)carve((size_t)G * DIM * 4);

  const int rowBlocks  = (N + 7) / 8;      // kernels with one wave per row
  const int edgeBlocks = (E + 7) / 8;      // one wave per edge
  const int mTiles     = (N + 15) / 16;    // WMMA stripes

  // Degree + normalization
  k_fill<<<(N + 255) / 256, 256, 0, stream>>>(deg, 1.0f, N);   // self-loop
  k_degree<<<(E + 255) / 256, 256, 0, stream>>>(dst, deg, E);
  k_dinv<<<(N + 255) / 256, 256, 0, stream>>>(deg, dinv, N);

  // 3 GCN layers
  const float* h_in = x;
  for (int l = 0; l < 3; ++l) {
    gemm_xw<<<mTiles, 256, 0, stream>>>(h_in, W[l], bufA, N);
    k_selfloop<<<rowBlocks, 256, 0, stream>>>(bufA, dinv, b[l], bufB, N);
    k_scatter<<<edgeBlocks, 256, 0, stream>>>(src, dst, dinv, bufA, bufB, E);
    k_ln_relu<<<rowBlocks, 256, 0, stream>>>(bufB, g[l], be[l], N);
    h_in = bufB;
  }

  // Global mean pool + FC + ReLU
  k_fill<<<(G * DIM + 255) / 256, 256, 0, stream>>>(sums, 0.0f, G * DIM);
  k_fill<<<(G + 255) / 256, 256, 0, stream>>>(cnt, 0.0f, G);
  k_pool<<<rowBlocks, 256, 0, stream>>>(bufB, batch, sums, cnt, N);
  k_pooled<<<(G * DIM + 255) / 256, 256, 0, stream>>>(sums, cnt, pooled, G * DIM);
  gemm_fc<<<(G + 15) / 16, 256, 0, stream>>>(pooled, fc_w, fc_b, out, G);
}